// MCSPACE_21285857919689
// MI455X (gfx1250) — compile-verified
//
#include <hip/hip_runtime.h>
#include <math.h>

// Problem dims (fixed by the reference)
#define TT 10
#define SSU 10
#define RR 2000
#define KK 100
#define OO 500
#define TS (TT * SSU)        // 100
#define KP 112               // K padded to 7*16 for WMMA N-tiles
#define NW 7                 // waves per block (one 16-col tile each)
#define NTHREADS (NW * 32)   // 224
#define OCHUNK (OO / 4)      // 125 WMMA K-steps (K=4 each)
#define MT (RR / 16)         // 125 row tiles per (t,s)
#define NWG (TS * MT)        // 12500 workgroups in main kernel
#define EPSF 1e-6f
#define PADNEG (-1.0e4f)     // rlog for pad cols ~ -5e7: never the max, exp -> 0

typedef float v2f __attribute__((ext_vector_type(2)));
typedef float v4f __attribute__((ext_vector_type(4)));
typedef float v8f __attribute__((ext_vector_type(8)));

// ds_swizzle_b32, group-of-32 mode: offset = {xor_mask[14:10], or=0, and=0x1f}
#define SWZ(v, imm) \
    __builtin_bit_cast(float, \
        __builtin_amdgcn_ds_swizzle(__builtin_bit_cast(int, (v)), (imm)))

// ---------------- Kernel 1: row softmax -> theta (out) + logtheta (ws) -----
__global__ __launch_bounds__(128)
void mcspace_softmax(const float* __restrict__ tp,
                     float* __restrict__ theta_out,
                     float* __restrict__ logtheta) {
    int k = blockIdx.x;
    const float* row = tp + (size_t)k * OO;
    __shared__ float sm[128];
    int t = threadIdx.x;

    float m = -3.4e38f;
    for (int o = t; o < OO; o += 128) m = fmaxf(m, row[o]);
    sm[t] = m;
    __syncthreads();
    for (int off = 64; off > 0; off >>= 1) {
        if (t < off) sm[t] = fmaxf(sm[t], sm[t + off]);
        __syncthreads();
    }
    float rowmax = sm[0];
    __syncthreads();

    float s = 0.f;
    for (int o = t; o < OO; o += 128) s += expf(row[o] - rowmax);
    sm[t] = s;
    __syncthreads();
    for (int off = 64; off > 0; off >>= 1) {
        if (t < off) sm[t] += sm[t + off];
        __syncthreads();
    }
    float inv = 1.0f / sm[0];

    for (int o = t; o < OO; o += 128) {
        float th = expf(row[o] - rowmax) * inv;
        theta_out[(size_t)k * OO + o] = th;
        logtheta[(size_t)k * OO + o] = logf(th + EPSF);
    }
}

// -------- Kernel 2: repack logtheta into WMMA-B swizzled layout -------------
// Bswz[((oc*KP) + n)*4 + j] = logtheta[n][oc*4 + j]  (n>=KK -> PADNEG)
__global__ __launch_bounds__(256)
void mcspace_bswz(const float* __restrict__ logtheta, float* __restrict__ Bswz) {
    int idx = blockIdx.x * 256 + threadIdx.x;
    if (idx >= OCHUNK * KP * 4) return;
    int j  = idx & 3;
    int n  = (idx >> 2) % KP;
    int oc = idx / (4 * KP);
    Bswz[idx] = (n < KK) ? logtheta[(size_t)n * OO + oc * 4 + j] : PADNEG;
}

// -------- Kernel 3: betaPad[ts][KP] (pad=0) + copy beta to output ----------
__global__ __launch_bounds__(256)
void mcspace_beta(const float* __restrict__ beta,
                  float* __restrict__ betaPad,
                  float* __restrict__ beta_out) {
    int idx = blockIdx.x * 256 + threadIdx.x;
    if (idx < TS * KP) {
        int ts = idx / KP;
        int k  = idx % KP;
        betaPad[idx] = (k < KK) ? beta[(size_t)k * TS + ts] : 0.f;
    }
    if (idx < KK * TS) beta_out[idx] = beta[idx];  // beta flat [K,T,S]
}

// -------- Kernel 4: fused GEMM (fp32 WMMA) + weighted logsumexp -------------
__global__ __launch_bounds__(NTHREADS)
void mcspace_main(const float* __restrict__ counts,
                  const float* __restrict__ Bswz,
                  const float* __restrict__ betaPad,
                  float* __restrict__ partials) {
    __shared__ __align__(16) float Atile[16 * OO];  // 32 KB: 16 rows x 500 cols
    __shared__ float red[16 * NW];                  // per-row, per-wave scratch
    __shared__ float gred[16];                      // global row max

    const int ts   = blockIdx.x / MT;
    const int mt   = blockIdx.x % MT;
    const int tid  = threadIdx.x;
    const int wave = tid >> 5;
    const int lane = tid & 31;
    const int l16  = lane & 15;
    const int half = lane >> 4;        // 0: K={0,1}, 1: K={2,3}

    // Cooperative 16B-vector, non-temporal load of the 16x500 counts tile.
    // counts is streamed exactly once -> NT keeps Bswz/betaPad hot in cache.
    const v4f* Asrc = (const v4f*)(counts + ((size_t)ts * RR + (size_t)mt * 16) * OO);
    v4f* Adst = (v4f*)Atile;
#pragma unroll
    for (int i = tid; i < (16 * OO) / 4; i += NTHREADS)
        Adst[i] = __builtin_nontemporal_load(Asrc + i);
    __syncthreads();

    // WMMA main loop: D(16x16) += A(16x4) * B(4x16) over 125 K-chunks
    v8f c = {};
    const float* Albase = Atile + (size_t)l16 * OO + 2 * half;
    const float* Bbase  = Bswz + ((size_t)(wave * 16 + l16)) * 4 + 2 * half;
    for (int oc = 0; oc < OCHUNK; ++oc) {
        v2f a = *(const v2f*)(Albase + oc * 4);                    // ds_load_b64
        v2f b = *(const v2f*)(Bbase + (size_t)oc * (KP * 4));      // global_load_b64
        c = __builtin_amdgcn_wmma_f32_16x16x4_f32(
                false, a, false, b, (short)0, c, false, false);
    }

    // --- per-wave row max over its 16 columns (ds_swizzle xor butterfly) ---
    // C/D layout: VGPR j, lanes 0-15 -> row j, col lane; lanes 16-31 -> row j+8
    float rmax[8];
#pragma unroll
    for (int j = 0; j < 8; ++j) {
        float v = c[j];
        v = fmaxf(v, SWZ(v, 0x041F));   // xor 1
        v = fmaxf(v, SWZ(v, 0x081F));   // xor 2
        v = fmaxf(v, SWZ(v, 0x101F));   // xor 4
        v = fmaxf(v, SWZ(v, 0x201F));   // xor 8 (stays within 16-lane half)
        rmax[j] = v;
    }
    if (l16 == 0) {
#pragma unroll
        for (int j = 0; j < 8; ++j)
            red[(j + 8 * half) * NW + wave] = rmax[j];
    }
    __syncthreads();

    // global row max across the 7 waves
    float gmax[8];
#pragma unroll
    for (int j = 0; j < 8; ++j) {
        int r = j + 8 * half;
        float mv = red[r * NW + 0];
#pragma unroll
        for (int w = 1; w < NW; ++w) mv = fmaxf(mv, red[r * NW + w]);
        gmax[j] = mv;
    }
    if (wave == 0 && l16 == 0) {
#pragma unroll
        for (int j = 0; j < 8; ++j) gred[j + 8 * half] = gmax[j];
    }
    __syncthreads();  // everyone done reading red before it is reused

    // --- per-wave partial sum of beta[k] * exp(rlog - rowmax) ---
    float bv = betaPad[(size_t)ts * KP + wave * 16 + l16];  // pad cols: beta=0
#pragma unroll
    for (int j = 0; j < 8; ++j) {
        float v = bv * expf(c[j] - gmax[j]);
        v += SWZ(v, 0x041F);
        v += SWZ(v, 0x081F);
        v += SWZ(v, 0x101F);
        v += SWZ(v, 0x201F);
        if (l16 == 0) red[(j + 8 * half) * NW + wave] = v;
    }
    __syncthreads();

    // one thread finalizes: summand[r] = rowmax[r] + log(sum + eps)
    if (tid == 0) {
        float acc = 0.f;
        for (int r = 0; r < 16; ++r) {
            float s = 0.f;
            for (int w = 0; w < NW; ++w) s += red[r * NW + w];
            acc += gred[r] + logf(s + EPSF);
        }
        partials[blockIdx.x] = acc;
    }
}

// -------- Kernel 5: deterministic fixed-order reduction -> -loglik ---------
__global__ __launch_bounds__(256)
void mcspace_reduce(const float* __restrict__ partials, float* __restrict__ out) {
    __shared__ float sm[256];
    int t = threadIdx.x;
    float acc = 0.f;
    for (int i = t; i < NWG; i += 256) acc += partials[i];
    sm[t] = acc;
    __syncthreads();
    for (int off = 128; off > 0; off >>= 1) {
        if (t < off) sm[t] += sm[t + off];
        __syncthreads();
    }
    if (t == 0) out[0] = -sm[0];
}

extern "C" void kernel_launch(void* const* d_in, const int* in_sizes, int n_in,
                              void* d_out, int out_size, void* d_ws, size_t ws_size,
                              hipStream_t stream) {
    const float* counts       = (const float*)d_in[0];  // [T,S,R,O]
    const float* theta_params = (const float*)d_in[1];  // [K,O]
    const float* beta         = (const float*)d_in[2];  // [K,T,S]
    float* out = (float*)d_out;   // [0]=elbo, [1..50001)=theta, [50001..60001)=beta
    float* ws  = (float*)d_ws;

    // Workspace layout (floats)
    float* logtheta = ws;                        // 50,000
    float* Bswz     = ws + 50000;                // 125*112*4 = 56,000
    float* betaPad  = ws + 106000;               // 100*112   = 11,200
    float* partials = ws + 117200;               // 12,500
    (void)in_sizes; (void)n_in; (void)out_size; (void)ws_size;

    mcspace_softmax<<<KK, 128, 0, stream>>>(theta_params, out + 1, logtheta);
    mcspace_bswz<<<(OCHUNK * KP * 4 + 255) / 256, 256, 0, stream>>>(logtheta, Bswz);
    mcspace_beta<<<(TS * KP + 255) / 256, 256, 0, stream>>>(beta, betaPad,
                                                            out + 1 + KK * OO);
    mcspace_main<<<NWG, NTHREADS, 0, stream>>>(counts, Bswz, betaPad, partials);
    mcspace_reduce<<<1, 256, 0, stream>>>(partials, out);
}